// NCC_31542239822445
// MI455X (gfx1250) — compile-verified
//
#include <hip/hip_runtime.h>

// ---------------------------------------------------------------------------
// Fused 3D local-NCC loss for 192^3 fp32 volumes, kernel_size=7.
// One workgroup = one 16^3 output tile. 22^3 halos of pred/targ live in LDS
// (CDNA5: 320 KB per WGP). The separable 7-wide box sums are computed as
// banded-matrix multiplies with V_WMMA_F32_16X16X4_F32 (full fp32 WMMA):
//     out(16 x N) = S(16 x 22) * in(22 x N),  S[i][k] = 1 if i <= k <= i+6
// chained over 6 K-chunks of 4. Three stages (z, y, x) per field, 5 fields
// (t, p, t^2, p^2, t*p) fully unrolled, then the pointwise cc ratio and a
// block reduction. All inner-loop masking is branchless (v_cndmask), no
// divergent stores.
// ---------------------------------------------------------------------------

typedef __attribute__((ext_vector_type(2))) float v2f;
typedef __attribute__((ext_vector_type(8))) float v8f;

#define NVOL   192
#define TILE   16
#define HALO   22            // TILE + 6
#define HALO2  484           // 22*22
#define HALOV  10648         // 22^3
#define ZSTR   496           // 31*16: padded column stride of sh_z
#define KSUM   343.0f        // 7^3
#define EPSN   1e-5f
#define EPSD   1e-5f

// LDS layout (floats)
#define SH_P    0                       // pred halo   [22][22][22]   10648
#define SH_T    10648                   // targ halo   [22][22][22]   10648
#define SH_Z    21296                   // z-filtered  [16][496]       7936
#define SH_ZY   29232                   // zy-filtered [16][352]       5632
#define SH_RES  34864                   // results     [5][4096]      20480
#define SH_TOT  55344                   // 221,376 bytes < 320 KB

// band(m,k): 1 if output row m's 7-window covers halo index k (k < 22)
__device__ __forceinline__ float band(int m, int k) {
    return ((unsigned)(k - m) <= 6u && k < HALO) ? 1.0f : 0.0f;
}

__global__ __launch_bounds__(256)
void ncc_tile_kernel(const float* __restrict__ pred,
                     const float* __restrict__ targ,
                     float* __restrict__ partial) {
    __shared__ float sh[SH_TOT];

    const int tid    = threadIdx.x;
    const int wave   = tid >> 5;        // 8 waves of 32 (wave32)
    const int lane   = tid & 31;
    const int lanelo = lane & 15;       // column index within tile / A row
    const int half   = lane >> 4;       // K-pair selector (0: K0/K1, 1: K2/K3)
    const int mrow   = lanelo;
    const int mbase  = half * 8;        // D-row base for this lane half

    // ---- load 22^3 halos of pred & targ (zero padding outside volume) ----
    const int zb = (int)blockIdx.z * TILE - 3;
    const int yb = (int)blockIdx.y * TILE - 3;
    const int xb = (int)blockIdx.x * TILE - 3;
    for (int i = tid; i < HALOV; i += 256) {
        int hz = i / HALO2; int r = i - hz * HALO2;
        int hy = r / HALO;  int hx = r - hy * HALO;
        int gz = zb + hz, gy = yb + hy, gx = xb + hx;
        bool in = ((unsigned)gz < (unsigned)NVOL) &
                  ((unsigned)gy < (unsigned)NVOL) &
                  ((unsigned)gx < (unsigned)NVOL);
        float pv = 0.0f, tv = 0.0f;
        if (in) {
            int gi = (gz * NVOL + gy) * NVOL + gx;
            pv = pred[gi];
            tv = targ[gi];
        }
        sh[SH_P + i] = pv;
        sh[SH_T + i] = tv;
    }

    #pragma unroll
    for (int f = 0; f < 5; ++f) {
        // ========== stage Z: M=oz(16), K=hz(22), N=(hy,hx) 484(+pad) ======
        __syncthreads();
        for (int t = wave; t < 31; t += 8) {
            const int n = t * 16 + lanelo;          // column in [0,496)
            v8f c = {};
            #pragma unroll
            for (int ck = 0; ck < 6; ++ck) {
                const int k0 = ck * 4 + half * 2;
                const int k1 = k0 + 1;
                v2f a, b;
                a.x = band(mrow, k0);
                a.y = band(mrow, k1);
                // clamped, always-valid addresses; value select for k>=22
                const int k0c = (k0 < HALO) ? k0 : 0;
                const int k1c = (k1 < HALO) ? k1 : 0;
                const int i0 = k0c * HALO2 + n;
                const int i1 = k1c * HALO2 + n;
                float bx, by;
                if (f == 0)      { bx = sh[SH_T + i0]; by = sh[SH_T + i1]; }
                else if (f == 1) { bx = sh[SH_P + i0]; by = sh[SH_P + i1]; }
                else if (f == 2) { float t0 = sh[SH_T + i0], t1 = sh[SH_T + i1];
                                   bx = t0 * t0; by = t1 * t1; }
                else if (f == 3) { float p0 = sh[SH_P + i0], p1 = sh[SH_P + i1];
                                   bx = p0 * p0; by = p1 * p1; }
                else             { bx = sh[SH_T + i0] * sh[SH_P + i0];
                                   by = sh[SH_T + i1] * sh[SH_P + i1]; }
                b.x = (k0 < HALO) ? bx : 0.0f;
                b.y = (k1 < HALO) ? by : 0.0f;
                c = __builtin_amdgcn_wmma_f32_16x16x4_f32(
                        false, a, false, b, (short)0, c, false, false);
            }
            #pragma unroll
            for (int v = 0; v < 8; ++v)             // unguarded: padded stride
                sh[SH_Z + (mbase + v) * ZSTR + n] = c[v];
        }

        // ========== stage Y: M=oy(16), K=hy(22), N=(oz,hx) 352 ============
        __syncthreads();
        for (int t = wave; t < 22; t += 8) {
            const int n  = t * 16 + lanelo;         // col = oz*22 + hx
            const int oz = n / HALO;
            const int hx = n - oz * HALO;
            const int zrow = SH_Z + oz * ZSTR + hx;
            v8f c = {};
            #pragma unroll
            for (int ck = 0; ck < 6; ++ck) {
                const int k0 = ck * 4 + half * 2;
                const int k1 = k0 + 1;
                const int k0c = (k0 < HALO) ? k0 : 0;
                const int k1c = (k1 < HALO) ? k1 : 0;
                v2f a, b;
                a.x = band(mrow, k0);
                a.y = band(mrow, k1);
                float bx = sh[zrow + k0c * HALO];
                float by = sh[zrow + k1c * HALO];
                b.x = (k0 < HALO) ? bx : 0.0f;
                b.y = (k1 < HALO) ? by : 0.0f;
                c = __builtin_amdgcn_wmma_f32_16x16x4_f32(
                        false, a, false, b, (short)0, c, false, false);
            }
            #pragma unroll
            for (int v = 0; v < 8; ++v)
                sh[SH_ZY + (mbase + v) * 352 + n] = c[v];
        }

        // ========== stage X: M=ox(16), K=hx(22), N=(oz,oy) 256 ============
        __syncthreads();
        for (int t = wave; t < 16; t += 8) {
            const int n  = t * 16 + lanelo;         // col = oz*16 + oy
            const int oz = n >> 4;
            const int oy = n & 15;
            const int zyrow = SH_ZY + oy * 352 + oz * HALO;
            v8f c = {};
            #pragma unroll
            for (int ck = 0; ck < 6; ++ck) {
                const int k0 = ck * 4 + half * 2;
                const int k1 = k0 + 1;
                const int k0c = (k0 < HALO) ? k0 : 0;
                const int k1c = (k1 < HALO) ? k1 : 0;
                v2f a, b;
                a.x = band(mrow, k0);
                a.y = band(mrow, k1);
                float bx = sh[zyrow + k0c];
                float by = sh[zyrow + k1c];
                b.x = (k0 < HALO) ? bx : 0.0f;
                b.y = (k1 < HALO) ? by : 0.0f;
                c = __builtin_amdgcn_wmma_f32_16x16x4_f32(
                        false, a, false, b, (short)0, c, false, false);
            }
            #pragma unroll
            for (int v = 0; v < 8; ++v)
                sh[SH_RES + f * 4096 + n * 16 + (mbase + v)] = c[v];
        }
    }

    // ================= pointwise cc + block reduction =====================
    __syncthreads();
    float acc = 0.0f;
    for (int i = tid; i < 4096; i += 256) {
        float ts = sh[SH_RES + i];
        float ps = sh[SH_RES + 4096  + i];
        float t2 = sh[SH_RES + 8192  + i];
        float p2 = sh[SH_RES + 12288 + i];
        float tp = sh[SH_RES + 16384 + i];
        float cross = tp - ts * ps / KSUM;
        float tv = fmaxf(t2 - ts * ts / KSUM, 0.0f);
        float pv = fmaxf(p2 - ps * ps / KSUM, 0.0f);
        acc += (cross * cross + EPSN) / (tv * pv + EPSD);
    }
    __syncthreads();
    sh[SH_Z + tid] = acc;
    __syncthreads();
    for (int off = 128; off > 0; off >>= 1) {
        if (tid < off) sh[SH_Z + tid] += sh[SH_Z + tid + off];
        __syncthreads();
    }
    if (tid == 0) {
        int bid = ((int)blockIdx.z * 12 + (int)blockIdx.y) * 12 + (int)blockIdx.x;
        partial[bid] = sh[SH_Z];
    }
}

__global__ __launch_bounds__(256)
void ncc_reduce_kernel(const float* __restrict__ partial,
                       float* __restrict__ out, int n) {
    __shared__ float s[256];
    float acc = 0.0f;
    for (int i = threadIdx.x; i < n; i += 256) acc += partial[i];
    s[threadIdx.x] = acc;
    __syncthreads();
    for (int off = 128; off > 0; off >>= 1) {
        if (threadIdx.x < off) s[threadIdx.x] += s[threadIdx.x + off];
        __syncthreads();
    }
    if (threadIdx.x == 0)
        out[0] = -s[0] / ((float)NVOL * (float)NVOL * (float)NVOL);
}

extern "C" void kernel_launch(void* const* d_in, const int* in_sizes, int n_in,
                              void* d_out, int out_size, void* d_ws, size_t ws_size,
                              hipStream_t stream) {
    const float* pred = (const float*)d_in[0];
    const float* targ = (const float*)d_in[1];
    float* partial = (float*)d_ws;          // 12*12*12 = 1728 floats

    dim3 grid(12, 12, 12);
    ncc_tile_kernel<<<grid, 256, 0, stream>>>(pred, targ, partial);
    ncc_reduce_kernel<<<1, 256, 0, stream>>>(partial, (float*)d_out, 12 * 12 * 12);
}